// SimpleRNN_84327387889756
// MI455X (gfx1250) — compile-verified
//
#include <hip/hip_runtime.h>

#define VOCAB 50257
#define EMB   50
#define HID   64
#define BATCH 256
#define SEQ   512
#define VT    3142   // ceil(VOCAB/16)

typedef __attribute__((ext_vector_type(2))) float v2f;
typedef __attribute__((ext_vector_type(8))) float v8f;

// D = A(16x4 f32) x B(4x16 f32) + C(16x16 f32), full f32 precision on the matrix pipe.
static __device__ __forceinline__ v8f wmma_f32(v2f a, v2f b, v8f c) {
  return __builtin_amdgcn_wmma_f32_16x16x4_f32(false, a, false, b, (short)0, c,
                                               false, false);
}

// tanh via v_exp_f32 + v_rcp_f32 on |x| (no overflow; sign restored at the end).
static __device__ __forceinline__ float fast_tanh(float x) {
  float ax = __builtin_fabsf(x);
  float e  = __expf(-2.0f * ax);
  float t  = (1.0f - e) * __builtin_amdgcn_rcpf(1.0f + e);
  return __builtin_copysignf(t, x);
}

// ---------------------------------------------------------------------------
// Kernel 1: embedding gather + input projection + biases, fused via WMMA.
// Output xh is stored pre-tiled in the WMMA C/D per-lane layout:
//   xh[(((s*16 + bt)*4 + jt)*32 + lane)*8 + r]
// so the recurrence kernel's accumulator init is two b128 loads per tile.
// ---------------------------------------------------------------------------
__global__ __launch_bounds__(256) void k1_embed_inproj(
    const int* __restrict__ x, const float* __restrict__ emb,
    const float* __restrict__ W_ih, const float* __restrict__ b_ih,
    const float* __restrict__ b_hh, float* __restrict__ xh) {
  const int lane = threadIdx.x & 31;
  const int row  = lane & 15;
  const int half = (lane >> 4) & 1;
  const int bt   = blockIdx.x * 8 + (threadIdx.x >> 5);  // 0..15
  const int s    = blockIdx.y;                           // 0..511

  const int b   = bt * 16 + row;
  const int idx = x[b * SEQ + s];
  const float* erow = emb + (size_t)idx * EMB;

  // A: embedded rows, K padded 50 -> 52 (13 chunks); branch-free clamp+mask.
  v2f a[13];
#pragma unroll
  for (int c = 0; c < 13; ++c) {
    int   k  = 4 * c + 2 * half;
    int   kk = k > 48 ? 48 : k;
    float m  = (k <= 48) ? 1.0f : 0.0f;
    v2f   w  = *(const v2f*)(erow + kk);
    a[c] = w * m;
  }

  // B: W_ih^T tiles (B[k][n] = W_ih[jt*16+n][k]).
  v2f bw[4][13];
#pragma unroll
  for (int jt = 0; jt < 4; ++jt) {
    const float* wrow = W_ih + (size_t)(jt * 16 + row) * EMB;
#pragma unroll
    for (int c = 0; c < 13; ++c) {
      int   k  = 4 * c + 2 * half;
      int   kk = k > 48 ? 48 : k;
      float m  = (k <= 48) ? 1.0f : 0.0f;
      v2f   w  = *(const v2f*)(wrow + kk);
      bw[jt][c] = w * m;
    }
  }

#pragma unroll
  for (int jt = 0; jt < 4; ++jt) {
    float bias = b_ih[jt * 16 + row] + b_hh[jt * 16 + row];
    v8f cc;
#pragma unroll
    for (int r = 0; r < 8; ++r) cc[r] = bias;
#pragma unroll
    for (int c = 0; c < 13; ++c) cc = wmma_f32(a[c], bw[jt][c], cc);
    float* outp = xh + ((((size_t)s * 16 + bt) * 4 + jt) * 32 + lane) * 8;
    *(v8f*)outp = cc;  // 2x global_store_b128, 32B aligned
  }
}

// ---------------------------------------------------------------------------
// Kernel 2: serial tanh recurrence (the latency-critical 512-step chain).
// 16 single-wave workgroups, each owning 16 batch rows for all 512 steps.
//  - W_hh^T resident in VGPRs as WMMA B-tiles (4 jt x 16 K-chunks).
//  - h round-trips through LDS only for the D-layout -> A-layout transpose.
//  - xh accumulator tiles double-buffered: step t+1's global loads issue at
//    the top of step t, hiding L2/HBM latency behind the WMMA chain.
//  - global_prefetch_b8 pulls xh 8 steps ahead into L2.
//  - No __syncthreads(): single-wave WG + in-order DS gives LDS ordering;
//    an empty asm memory clobber pins compiler ordering without forcing a
//    loadcnt flush (which would defeat the software pipeline).
// ---------------------------------------------------------------------------
__global__ __launch_bounds__(32) void k2_recurrence(
    const float* __restrict__ W_hh, const float* __restrict__ xh,
    float* __restrict__ hfin) {
  const int lane = threadIdx.x;
  const int row  = lane & 15;
  const int half = lane >> 4;
  const int bt   = blockIdx.x;

  __shared__ float hbuf[16 * 68];  // [16 rows][64 cols], stride 68: conflict-free

  // W_hh^T resident in registers: B[k][n] = W_hh[jt*16+n][k]; 4 jt x 16 chunks.
  v2f bw[4][16];
#pragma unroll
  for (int jt = 0; jt < 4; ++jt) {
    const float* wrow = W_hh + (size_t)(jt * 16 + row) * HID;
#pragma unroll
    for (int c = 0; c < 16; ++c) bw[jt][c] = *(const v2f*)(wrow + 4 * c + 2 * half);
  }

  for (int i = lane; i < 16 * 68; i += 32) hbuf[i] = 0.0f;  // h0 = 0
  asm volatile("" ::: "memory");  // keep init stores before first A loads

  // xh flat index: t*16384 + bt*1024 + jt*256 + lane*8
  const float* xbase = xh + (size_t)bt * 1024 + lane * 8;

  v8f cbuf[4];
#pragma unroll
  for (int jt = 0; jt < 4; ++jt) cbuf[jt] = *(const v8f*)(xbase + jt * 256);

  for (int t = 0; t < SEQ; ++t) {
    // L2 prefetch of the xh stream, 8 steps ahead (uniform branch, EXEC intact)
    if (t + 8 < SEQ)
      __builtin_prefetch(xbase + (size_t)(t + 8) * 16384, 0, 3);

    // A = h (row in lane, K along vgprs): 16 ds_load_b64
    v2f a[16];
#pragma unroll
    for (int c = 0; c < 16; ++c)
      a[c] = *(const v2f*)&hbuf[row * 68 + 4 * c + 2 * half];

    // Issue next step's accumulator tiles now; consumed next iteration.
    const int tn = (t + 1 < SEQ) ? (t + 1) : (SEQ - 1);
    const float* xpn = xbase + (size_t)tn * 16384;
    v8f cnext[4];
#pragma unroll
    for (int jt = 0; jt < 4; ++jt) cnext[jt] = *(const v8f*)(xpn + jt * 256);

#pragma unroll
    for (int jt = 0; jt < 4; ++jt) {
      v8f cc = cbuf[jt];
#pragma unroll
      for (int c = 0; c < 16; ++c) cc = wmma_f32(a[c], bw[jt][c], cc);
      // tanh + transpose store (D: col in lane / row in vgpr -> LDS [row][col])
#pragma unroll
      for (int r = 0; r < 8; ++r)
        hbuf[(half * 8 + r) * 68 + jt * 16 + row] = fast_tanh(cc[r]);
    }

#pragma unroll
    for (int jt = 0; jt < 4; ++jt) cbuf[jt] = cnext[jt];
    asm volatile("" ::: "memory");  // order tanh stores before next-step A loads
  }

  for (int i = lane; i < 16 * HID; i += 32) {
    int m = i >> 6, k = i & 63;
    hfin[(size_t)(bt * 16 + m) * HID + k] = hbuf[m * 68 + k];
  }
}

// ---------------------------------------------------------------------------
// Kernel 3: head GEMM [256,64] x [64,50257] + bias. One wave per 16x16 output
// tile, K=64 -> 16 chained f32 WMMAs. Vocab edge handled branch-free so EXEC
// stays all-1s through the WMMAs.
// ---------------------------------------------------------------------------
__global__ __launch_bounds__(256) void k3_head(
    const float* __restrict__ hfin, const float* __restrict__ W_fc,
    const float* __restrict__ b_fc, float* __restrict__ out) {
  const int lane = threadIdx.x & 31;
  const int row  = lane & 15;
  const int half = (lane >> 4) & 1;
  const int vt   = blockIdx.x * 8 + (threadIdx.x >> 5);  // may exceed 3141; masked
  const int bt   = blockIdx.y;

  const int   j   = vt * 16 + row;
  const int   jc  = j < VOCAB ? j : (VOCAB - 1);
  const float msk = (j < VOCAB) ? 1.0f : 0.0f;

  v2f a[16], bw[16];
  const float* arow = hfin + (size_t)(bt * 16 + row) * HID;
  const float* wrow = W_fc + (size_t)jc * HID;
#pragma unroll
  for (int c = 0; c < 16; ++c) {
    int k = 4 * c + 2 * half;
    a[c]  = *(const v2f*)(arow + k);
    v2f w = *(const v2f*)(wrow + k);
    bw[c] = w * msk;
  }

  float bias = b_fc[jc] * msk;
  v8f cc;
#pragma unroll
  for (int r = 0; r < 8; ++r) cc[r] = bias;
#pragma unroll
  for (int c = 0; c < 16; ++c) cc = wmma_f32(a[c], bw[c], cc);

  if (j < VOCAB) {
#pragma unroll
    for (int r = 0; r < 8; ++r) {
      int m = half * 8 + r;
      out[(size_t)(bt * 16 + m) * VOCAB + j] = cc[r];
    }
  }
}

extern "C" void kernel_launch(void* const* d_in, const int* in_sizes, int n_in,
                              void* d_out, int out_size, void* d_ws, size_t ws_size,
                              hipStream_t stream) {
  const int*   x    = (const int*)d_in[0];
  const float* emb  = (const float*)d_in[1];
  const float* W_ih = (const float*)d_in[2];
  const float* b_ih = (const float*)d_in[3];
  const float* W_hh = (const float*)d_in[4];
  const float* b_hh = (const float*)d_in[5];
  const float* W_fc = (const float*)d_in[6];
  const float* b_fc = (const float*)d_in[7];
  float*       out  = (float*)d_out;

  float* xh   = (float*)d_ws;                   // B*S*HID f32 = 33.5 MB, WMMA-tiled
  float* hfin = xh + (size_t)BATCH * SEQ * HID; // B*HID f32

  k1_embed_inproj<<<dim3(2, SEQ), 256, 0, stream>>>(x, emb, W_ih, b_ih, b_hh, xh);
  k2_recurrence<<<dim3(16), 32, 0, stream>>>(W_hh, xh, hfin);
  k3_head<<<dim3((VT + 7) / 8, 16), 256, 0, stream>>>(hfin, W_fc, b_fc, out);
}